// Mamba_Block_82635170775464
// MI455X (gfx1250) — compile-verified
//
#include <hip/hip_runtime.h>
#include <hip/hip_bf16.h>

// ---------------------------------------------------------------------------
// Mamba block forward for MI455X (gfx1250, wave32).
// fp32 GEMMs via V_WMMA_F32_16X16X4_F32; B tiles staged through LDS with
// double-buffered async global->LDS copies (ASYNCcnt).
// ---------------------------------------------------------------------------

#define BB   2
#define LL   1024
#define DD   1024
#define DIN  2048
#define NS   16     // d_state
#define RR   64     // dt_rank
#define KC   4      // conv kernel
#define MROWS (BB * LL)          // 2048
#define XPC  (RR + 2 * NS)       // 96 columns of x_dbl

typedef float v2f __attribute__((ext_vector_type(2)));
typedef float v8f __attribute__((ext_vector_type(8)));

#ifndef __has_builtin
#define __has_builtin(x) 0
#endif

typedef __attribute__((address_space(1))) int* gptr_t;
typedef __attribute__((address_space(3))) int* lptr_t;

// ---- async global -> LDS copy (gfx1250), with safe sync fallback ----------
__device__ __forceinline__ void async_cp_b32(const float* gp, float* lp) {
#if __has_builtin(__builtin_amdgcn_global_load_async_to_lds_b32)
  __builtin_amdgcn_global_load_async_to_lds_b32((gptr_t)gp, (lptr_t)lp, 0, 0);
#else
  *lp = *gp;
#endif
}

__device__ __forceinline__ void wait_async0() {
#if __has_builtin(__builtin_amdgcn_global_load_async_to_lds_b32)
#if __has_builtin(__builtin_amdgcn_s_wait_asynccnt)
  __builtin_amdgcn_s_wait_asynccnt(0);
#else
  asm volatile("s_wait_asynccnt 0x0" ::: "memory");
#endif
#endif
}

// ---------------------------------------------------------------------------
// Cooperative stage of B[k0:k0+KB][nBase:nBase+BN] into LDS, pair-interleaved:
// lds pair p = ((kk/2)*BN + nn) holds (B[kk][nn], B[kk+1][nn]) for even kk,
// so the reader fetches one ds_load_b64 per WMMA fragment.
// ---------------------------------------------------------------------------
template <int BN, int KB>
__device__ __forceinline__ void stage_B(const float* __restrict__ Bm, int ldb,
                                        int k0, int nBase, float* lbase) {
  constexpr int PER_T = (KB * BN) / 256;
  const int tid = threadIdx.x;
#pragma unroll
  for (int j = 0; j < PER_T; ++j) {
    const int e  = tid + 256 * j;
    const int kk = e / BN;
    const int nn = e % BN;
    const float* gp = Bm + (size_t)(k0 + kk) * (size_t)ldb + nBase + nn;
    float* lp = lbase + ((kk >> 1) * BN + nn) * 2 + (kk & 1);
    async_cp_b32(gp, lp);
  }
}

// ---------------------------------------------------------------------------
// Block GEMM: 8 waves, each wave computes a 16(M) x (16*NT)(N) strip.
// A fragments loaded directly (per-wave unique rows); B fragments from LDS.
// WMMA f32 16x16x4 layout (ISA 7.12.2): lane t -> idx=t&15, h=t>>4
//   A[m=idx][k+2h+v] (contiguous v2f), B[k+2h+v][n=idx], D vgpr r=(m=r+8h,n=idx)
// ---------------------------------------------------------------------------
template <int KDIM, int NT, int KB>
__device__ __forceinline__ void gemm_block(const float* __restrict__ A, int lda,
                                           const float* __restrict__ Bm, int ldb,
                                           int mBlock, int nBase, v8f (&acc)[NT]) {
  constexpr int BN = NT * 16;
  constexpr int CHUNKS = KDIM / KB;
  __shared__ v2f ldsB[2][(KB / 2) * BN];

  const int tid  = threadIdx.x;
  const int wave = tid >> 5;
  const int lane = tid & 31;
  const int idx  = lane & 15;
  const int half = lane >> 4;
  const float* arow = A + (size_t)(mBlock + wave * 16 + idx) * (size_t)lda;

  stage_B<BN, KB>(Bm, ldb, 0, nBase, (float*)&ldsB[0][0]);

  for (int c = 0; c < CHUNKS; ++c) {
    wait_async0();
    __syncthreads();
    if (c + 1 < CHUNKS)
      stage_B<BN, KB>(Bm, ldb, (c + 1) * KB, nBase, (float*)&ldsB[(c + 1) & 1][0]);
    const v2f* lb = &ldsB[c & 1][0];
#pragma unroll
    for (int kk = 0; kk < KB; kk += 4) {
      const int kg = c * KB + kk;
      v2f a = *(const v2f*)(arow + kg + 2 * half);
      const v2f* lrow = lb + (((kk + 2 * half) >> 1) * BN + idx);
#pragma unroll
      for (int t = 0; t < NT; ++t) {
        v2f b = lrow[t * 16];
        acc[t] = __builtin_amdgcn_wmma_f32_16x16x4_f32(
            false, a, false, b, (short)0, acc[t], false, false);
      }
    }
    __syncthreads();
  }
}

__device__ __forceinline__ float silu_f(float v) {
  return v / (1.0f + __expf(-v));
}

// ---------------------------------------------------------------------------
// GEMM 1: xz = x[2048,1024] @ W_in[1024,4096]
// epilogue: n < DIN -> x_in raw ; n >= DIN -> silu(z)
// ---------------------------------------------------------------------------
__global__ __launch_bounds__(256) void k_gemm_in(const float* __restrict__ x,
                                                 const float* __restrict__ Wi,
                                                 float* __restrict__ xin,
                                                 float* __restrict__ zs) {
  const int mBlock = blockIdx.y * 128;
  const int nBase  = blockIdx.x * 64;
  v8f acc[4] = {};
  gemm_block<DD, 4, 32>(x, DD, Wi, 2 * DIN, mBlock, nBase, acc);
  const int wave = threadIdx.x >> 5;
  const int lane = threadIdx.x & 31;
  const int idx  = lane & 15;
  const int half = lane >> 4;
  const int mBase = mBlock + wave * 16;
#pragma unroll
  for (int t = 0; t < 4; ++t) {
    const int n = nBase + t * 16 + idx;
#pragma unroll
    for (int r = 0; r < 8; ++r) {
      const int m = mBase + r + 8 * half;
      const float v = acc[t][r];
      if (n < DIN) xin[(size_t)m * DIN + n] = v;
      else         zs[(size_t)m * DIN + (n - DIN)] = silu_f(v);
    }
  }
}

// ---------------------------------------------------------------------------
// Causal depthwise conv1d (K=4) + SiLU, elementwise over B*L*DIN
// ---------------------------------------------------------------------------
__global__ __launch_bounds__(256) void k_conv(const float* __restrict__ xin,
                                              const float* __restrict__ cw,
                                              const float* __restrict__ cb,
                                              float* __restrict__ xconv) {
  const int tid = blockIdx.x * 256 + threadIdx.x;
  if (tid >= BB * LL * DIN) return;
  const int d  = tid % DIN;
  const int bl = tid / DIN;
  const int l  = bl % LL;
  float acc = cb[d];
#pragma unroll
  for (int k = 0; k < KC; ++k) {
    const int ls = l - (KC - 1) + k;
    if (ls >= 0)
      acc += xin[(size_t)(bl - l + ls) * DIN + d] * cw[d * KC + k];
  }
  xconv[tid] = silu_f(acc);
}

// ---------------------------------------------------------------------------
// GEMM 2: x_dbl = x_conv[2048,2048] @ W_x[2048,96]
// ---------------------------------------------------------------------------
__global__ __launch_bounds__(256) void k_gemm_xproj(const float* __restrict__ xc,
                                                    const float* __restrict__ Wx,
                                                    float* __restrict__ xdbl) {
  const int mBlock = blockIdx.x * 128;
  v8f acc[6] = {};
  gemm_block<DIN, 6, 32>(xc, DIN, Wx, XPC, mBlock, 0, acc);
  const int wave = threadIdx.x >> 5;
  const int lane = threadIdx.x & 31;
  const int idx  = lane & 15;
  const int half = lane >> 4;
  const int mBase = mBlock + wave * 16;
#pragma unroll
  for (int t = 0; t < 6; ++t) {
    const int n = t * 16 + idx;
#pragma unroll
    for (int r = 0; r < 8; ++r)
      xdbl[(size_t)(mBase + r + 8 * half) * XPC + n] = acc[t][r];
  }
}

// ---------------------------------------------------------------------------
// GEMM 3: delta = softplus(dt[2048,64] @ W_dt[64,2048] + b_dt)
// dt = x_dbl[:, 0:64] (row stride 96)
// ---------------------------------------------------------------------------
__global__ __launch_bounds__(256) void k_gemm_dt(const float* __restrict__ xdbl,
                                                 const float* __restrict__ Wdt,
                                                 const float* __restrict__ bdt,
                                                 float* __restrict__ delta) {
  const int mBlock = blockIdx.y * 128;
  const int nBase  = blockIdx.x * 64;
  v8f acc[4] = {};
  gemm_block<RR, 4, 32>(xdbl, XPC, Wdt, DIN, mBlock, nBase, acc);
  const int wave = threadIdx.x >> 5;
  const int lane = threadIdx.x & 31;
  const int idx  = lane & 15;
  const int half = lane >> 4;
  const int mBase = mBlock + wave * 16;
#pragma unroll
  for (int t = 0; t < 4; ++t) {
    const int n = nBase + t * 16 + idx;
    const float bias = bdt[n];
#pragma unroll
    for (int r = 0; r < 8; ++r) {
      float v = acc[t][r] + bias;
      float sp = (v > 20.0f) ? v : log1pf(__expf(v));
      delta[(size_t)(mBase + r + 8 * half) * DIN + n] = sp;
    }
  }
}

// ---------------------------------------------------------------------------
// Selective scan: one thread per (b, d). h[16] in registers; B/C tiles staged
// through LDS in chunks of 64 timesteps. Fused D-skip + z-gating; output is
// written in place over the silu(z) buffer.
// ---------------------------------------------------------------------------
__global__ __launch_bounds__(256) void k_scan(const float* __restrict__ delta,
                                              const float* __restrict__ xconv,
                                              const float* __restrict__ xdbl,
                                              const float* __restrict__ A_log,
                                              const float* __restrict__ Dsk,
                                              float* __restrict__ yz) {
  __shared__ float sB[64 * NS];
  __shared__ float sC[64 * NS];
  const int b = blockIdx.y;
  const int d = blockIdx.x * 256 + threadIdx.x;
  float An[NS];
#pragma unroll
  for (int n = 0; n < NS; ++n) An[n] = -__expf(A_log[d * NS + n]);
  const float dskip = Dsk[d];
  float h[NS];
#pragma unroll
  for (int n = 0; n < NS; ++n) h[n] = 0.0f;
  const size_t rowb = (size_t)b * LL;

  for (int l0 = 0; l0 < LL; l0 += 64) {
    __syncthreads();
#pragma unroll
    for (int j = 0; j < 4; ++j) {
      const int e  = threadIdx.x + 256 * j;   // 1024 elements = 64 x 16
      const int li = e >> 4;
      const int n  = e & 15;
      const size_t base = (rowb + l0 + li) * (size_t)XPC;
      sB[e] = xdbl[base + RR + n];
      sC[e] = xdbl[base + RR + NS + n];
    }
    __syncthreads();
    for (int i = 0; i < 64; ++i) {
      const size_t off = (rowb + l0 + i) * (size_t)DIN + d;
      const float dlt = delta[off];
      const float xc  = xconv[off];
      const float zv  = yz[off];
      float y = 0.0f;
#pragma unroll
      for (int n = 0; n < NS; ++n) {
        const float dA = __expf(dlt * An[n]);
        h[n] = dA * h[n] + (dlt * sB[i * NS + n]) * xc;
        y += h[n] * sC[i * NS + n];
      }
      yz[off] = (y + xc * dskip) * zv;
    }
  }
}

// ---------------------------------------------------------------------------
// GEMM 4: pre = y_gated[2048,2048] @ W_out[2048,1024]
// ---------------------------------------------------------------------------
__global__ __launch_bounds__(256) void k_gemm_out(const float* __restrict__ yg,
                                                  const float* __restrict__ Wo,
                                                  float* __restrict__ pre) {
  const int mBlock = blockIdx.y * 128;
  const int nBase  = blockIdx.x * 64;
  v8f acc[4] = {};
  gemm_block<DIN, 4, 32>(yg, DIN, Wo, DD, mBlock, nBase, acc);
  const int wave = threadIdx.x >> 5;
  const int lane = threadIdx.x & 31;
  const int idx  = lane & 15;
  const int half = lane >> 4;
  const int mBase = mBlock + wave * 16;
#pragma unroll
  for (int t = 0; t < 4; ++t) {
    const int n = nBase + t * 16 + idx;
#pragma unroll
    for (int r = 0; r < 8; ++r)
      pre[(size_t)(mBase + r + 8 * half) * DD + n] = acc[t][r];
  }
}

// ---------------------------------------------------------------------------
// Row LayerNorm over D=1024, one block (256 threads, 4 elems each) per row.
// ---------------------------------------------------------------------------
__global__ __launch_bounds__(256) void k_ln(const float* __restrict__ pre,
                                            const float* __restrict__ g,
                                            const float* __restrict__ bta,
                                            float* __restrict__ out) {
  __shared__ float sh[256];
  __shared__ float sh2[256];
  const int row = blockIdx.x;
  const float* p = pre + (size_t)row * DD;
  float v[4];
  float s = 0.0f, s2 = 0.0f;
#pragma unroll
  for (int j = 0; j < 4; ++j) {
    v[j] = p[threadIdx.x + 256 * j];
    s += v[j];
    s2 += v[j] * v[j];
  }
  sh[threadIdx.x] = s;
  sh2[threadIdx.x] = s2;
  __syncthreads();
  for (int st = 128; st > 0; st >>= 1) {
    if (threadIdx.x < st) {
      sh[threadIdx.x] += sh[threadIdx.x + st];
      sh2[threadIdx.x] += sh2[threadIdx.x + st];
    }
    __syncthreads();
  }
  const float mu  = sh[0] * (1.0f / DD);
  const float var = sh2[0] * (1.0f / DD) - mu * mu;
  const float inv = rsqrtf(var + 1e-5f);
#pragma unroll
  for (int j = 0; j < 4; ++j) {
    const int n = threadIdx.x + 256 * j;
    out[(size_t)row * DD + n] = (v[j] - mu) * inv * g[n] + bta[n];
  }
}

// ---------------------------------------------------------------------------
extern "C" void kernel_launch(void* const* d_in, const int* in_sizes, int n_in,
                              void* d_out, int out_size, void* d_ws, size_t ws_size,
                              hipStream_t stream) {
  const float* x      = (const float*)d_in[0];
  const float* W_in   = (const float*)d_in[1];
  const float* conv_w = (const float*)d_in[2];
  const float* conv_b = (const float*)d_in[3];
  const float* W_x    = (const float*)d_in[4];
  const float* W_dt   = (const float*)d_in[5];
  const float* b_dt   = (const float*)d_in[6];
  const float* A_log  = (const float*)d_in[7];
  const float* D_skip = (const float*)d_in[8];
  const float* W_out  = (const float*)d_in[9];
  const float* ln_g   = (const float*)d_in[10];
  const float* ln_b   = (const float*)d_in[11];
  float* out = (float*)d_out;

  // workspace partition (floats):
  //  buf0: x_in  -> delta -> preLN   (MROWS*DIN)
  //  buf1: silu(z) -> y_gated (in place)
  //  buf2: x_conv
  //  xdbl: [MROWS, 96]
  float* buf0 = (float*)d_ws;
  float* buf1 = buf0 + (size_t)MROWS * DIN;
  float* buf2 = buf1 + (size_t)MROWS * DIN;
  float* xdbl = buf2 + (size_t)MROWS * DIN;

  // 1) in_proj (split epilogue: x_in raw, silu(z))
  k_gemm_in<<<dim3((2 * DIN) / 64, MROWS / 128), 256, 0, stream>>>(x, W_in, buf0, buf1);
  // 2) causal depthwise conv + silu
  k_conv<<<(BB * LL * DIN + 255) / 256, 256, 0, stream>>>(buf0, conv_w, conv_b, buf2);
  // 3) x_proj -> dt|B|C
  k_gemm_xproj<<<MROWS / 128, 256, 0, stream>>>(buf2, W_x, xdbl);
  // 4) delta = softplus(dt @ W_dt + b_dt)   (overwrites x_in)
  k_gemm_dt<<<dim3(DIN / 64, MROWS / 128), 256, 0, stream>>>(xdbl, W_dt, b_dt, buf0);
  // 5) selective scan + D-skip + z-gate (in place over buf1)
  k_scan<<<dim3(DIN / 256, BB), 256, 0, stream>>>(buf0, buf2, xdbl, A_log, D_skip, buf1);
  // 6) out_proj (overwrites delta)
  k_gemm_out<<<dim3(DD / 64, MROWS / 128), 256, 0, stream>>>(buf1, W_out, buf0);
  // 7) layernorm -> d_out
  k_ln<<<MROWS, 256, 0, stream>>>(buf0, ln_g, ln_b, out);
}